// GCN_360777253171
// MI455X (gfx1250) — compile-verified
//
#include <hip/hip_runtime.h>
#include <hip/hip_bf16.h>

typedef __attribute__((ext_vector_type(2))) float v2f;
typedef __attribute__((ext_vector_type(8))) float v8f;

#define N_IN  128
#define N_HID 128
#define N_OUT 64

// ---------------- degree kernels ----------------
__global__ void k_deg_init(float* __restrict__ deg, int n) {
  int i = blockIdx.x * blockDim.x + threadIdx.x;
  if (i < n) deg[i] = 1.0f;  // self-loop contributes 1
}

__global__ void k_deg_count(float* __restrict__ deg, const int* __restrict__ dst, int E) {
  int e = blockIdx.x * blockDim.x + threadIdx.x;
  if (e < E) unsafeAtomicAdd(&deg[dst[e]], 1.0f);
}

__global__ void k_rsqrt_inplace(float* __restrict__ d, int n) {
  int i = blockIdx.x * blockDim.x + threadIdx.x;
  if (i < n) d[i] = rsqrtf(d[i]);
}

// ---------------- WMMA GEMM: C[M x N] = A[M x 128] * B[128 x N] ----------------
// One block = one 16-row M tile; one wave per 16-col N tile. K fixed at 128.
// A-frag layout (32-bit A 16x4, ISA 7.12.2): lanes 0-15 M=lane, VGPR0=K+0,VGPR1=K+1;
// lanes 16-31 M=lane-16, VGPR0=K+2, VGPR1=K+3.  B mirrors with N=lane%16.
// C/D: VGPR r -> M=r (lanes 0-15) / M=r+8 (lanes 16-31), N=lane%16.
template <int N>
__global__ __launch_bounds__(32 * (N / 16)) void k_gemm_wmma(
    const float* __restrict__ A, const float* __restrict__ B, float* __restrict__ C) {
  constexpr int K = 128;
  constexpr int LDA = K + 4;  // pad: bank-conflict-free frag reads (stride 132 words)
  __shared__ float As[16 * LDA];

  const int tid = threadIdx.x;
  const size_t tileM = blockIdx.x;

  // 16xK tile of A is contiguous (row stride == K); coalesced stage into LDS.
  const float* Ag = A + tileM * (size_t)(16 * K);
  for (int i = tid; i < 16 * K; i += 32 * (N / 16)) {
    As[(i / K) * LDA + (i % K)] = Ag[i];
  }
  __syncthreads();

  const int lane = tid & 31;
  const int wave = tid >> 5;
  const int m  = lane & 15;
  const int hi = lane >> 4;
  const int nc = (wave << 4) + m;  // output column for B/C frags

  v8f acc = {};
#pragma unroll
  for (int k = 0; k < K; k += 4) {
    const int kk = k + (hi << 1);
    v2f a;
    a.x = As[m * LDA + kk];
    a.y = As[m * LDA + kk + 1];
    v2f b;
    b.x = B[(size_t)kk * N + nc];
    b.y = B[(size_t)(kk + 1) * N + nc];
    // (neg_a, A, neg_b, B, c_mod, C, reuse_a, reuse_b)
    acc = __builtin_amdgcn_wmma_f32_16x16x4_f32(false, a, false, b, (short)0, acc,
                                                false, false);
  }

  float* Cp = C + (tileM * 16 + (hi << 3)) * (size_t)N + (wave << 4) + m;
#pragma unroll
  for (int r = 0; r < 8; ++r) Cp[(size_t)r * N] = acc[r];
}

// ---------------- self-loop init: agg[i][c] = H[i][c] * dis[i]^2 ----------------
template <int F>
__global__ void k_self_init(float* __restrict__ agg, const float* __restrict__ H,
                            const float* __restrict__ dis, int n) {
  long long idx = (long long)blockIdx.x * blockDim.x + threadIdx.x;
  if (idx >= (long long)n * F) return;
  int i = (int)(idx / F);
  float w = dis[i];
  agg[idx] = H[idx] * (w * w);
}

// ---------------- edge scatter: agg[dst] += H[src] * dis[src]*dis[dst] ----------------
// Thread handles one float4 chunk of one edge; consecutive threads -> consecutive
// chunks of the same edge (coalesced gather + coalesced atomic targets).
template <int F>
__global__ void k_scatter(float* __restrict__ agg, const float* __restrict__ H,
                          const float* __restrict__ dis, const int* __restrict__ src,
                          const int* __restrict__ dst, int E) {
  constexpr int CH = F / 4;
  long long idx = (long long)blockIdx.x * blockDim.x + threadIdx.x;
  if (idx >= (long long)E * CH) return;
  int e = (int)(idx / CH);
  int c = (int)(idx % CH);
  int s = src[e];
  int d = dst[e];
  float nrm = dis[s] * dis[d];
  float4 h = ((const float4*)(H + (size_t)s * F))[c];
  float* p = agg + (size_t)d * F + 4 * c;
  unsafeAtomicAdd(p + 0, h.x * nrm);
  unsafeAtomicAdd(p + 1, h.y * nrm);
  unsafeAtomicAdd(p + 2, h.z * nrm);
  unsafeAtomicAdd(p + 3, h.w * nrm);
}

// ---------------- bias + ReLU (in place) ----------------
template <int F>
__global__ void k_bias_relu(float* __restrict__ a, const float* __restrict__ b, int n) {
  long long idx = (long long)blockIdx.x * blockDim.x + threadIdx.x;
  if (idx >= (long long)n * F) return;
  float v = a[idx] + b[(int)(idx % F)];
  a[idx] = v > 0.0f ? v : 0.0f;
}

// ---------------- bias + log_softmax over 64 channels, one wave32 per node ----------------
__global__ __launch_bounds__(256) void k_logsoftmax64(const float* __restrict__ agg,
                                                      const float* __restrict__ b,
                                                      float* __restrict__ out, int n) {
  int node = blockIdx.x * (blockDim.x >> 5) + (threadIdx.x >> 5);
  if (node >= n) return;
  int lane = threadIdx.x & 31;
  size_t base = (size_t)node * 64;
  float v0 = agg[base + lane] + b[lane];
  float v1 = agg[base + 32 + lane] + b[32 + lane];
  float m = fmaxf(v0, v1);
#pragma unroll
  for (int o = 16; o > 0; o >>= 1) m = fmaxf(m, __shfl_xor(m, o, 32));
  float s = expf(v0 - m) + expf(v1 - m);
#pragma unroll
  for (int o = 16; o > 0; o >>= 1) s += __shfl_xor(s, o, 32);
  float lse = logf(s) + m;
  out[base + lane] = v0 - lse;
  out[base + 32 + lane] = v1 - lse;
}

// ---------------- launch ----------------
extern "C" void kernel_launch(void* const* d_in, const int* in_sizes, int n_in,
                              void* d_out, int out_size, void* d_ws, size_t ws_size,
                              hipStream_t stream) {
  const float* x  = (const float*)d_in[0];
  const int*   ei = (const int*)d_in[1];
  const float* W1 = (const float*)d_in[2];
  const float* b1 = (const float*)d_in[3];
  const float* W2 = (const float*)d_in[4];
  const float* b2 = (const float*)d_in[5];

  const int n = in_sizes[0] / N_IN;   // 100000 (divisible by 16)
  const int E = in_sizes[1] / 2;      // 1600000
  const int* srcv = ei;
  const int* dstv = ei + E;

  // workspace layout
  char* ws = (char*)d_ws;
  size_t off = 0;
  auto alloc = [&](size_t bytes) -> void* {
    void* p = ws + off;
    off += bytes;
    off = (off + 255) & ~(size_t)255;
    return p;
  };
  float* dis = (float*)alloc((size_t)n * sizeof(float));
  float* H   = (float*)alloc((size_t)n * N_HID * sizeof(float));  // H1, later H2
  float* AG  = (float*)alloc((size_t)n * N_HID * sizeof(float));  // agg1, later agg2
  float* out = (float*)d_out;

  const int T = 256;
  auto blks = [&](long long total) { return (int)((total + T - 1) / T); };

  // degrees -> dis = rsqrt(deg)
  k_deg_init<<<blks(n), T, 0, stream>>>(dis, n);
  k_deg_count<<<blks(E), T, 0, stream>>>(dis, dstv, E);
  k_rsqrt_inplace<<<blks(n), T, 0, stream>>>(dis, n);

  // layer 1: H = x @ W1 ; agg1 = scatter(norm * H) ; relu(agg1 + b1)
  k_gemm_wmma<N_HID><<<n / 16, 256, 0, stream>>>(x, W1, H);
  k_self_init<N_HID><<<blks((long long)n * N_HID), T, 0, stream>>>(AG, H, dis, n);
  k_scatter<N_HID><<<blks((long long)E * (N_HID / 4)), T, 0, stream>>>(AG, H, dis, srcv,
                                                                       dstv, E);
  k_bias_relu<N_HID><<<blks((long long)n * N_HID), T, 0, stream>>>(AG, b1, n);

  // layer 2: H2 = agg1 @ W2 (reuse H buffer) ; agg2 = scatter (reuse AG buffer)
  k_gemm_wmma<N_OUT><<<n / 16, 128, 0, stream>>>(AG, W2, H);
  k_self_init<N_OUT><<<blks((long long)n * N_OUT), T, 0, stream>>>(AG, H, dis, n);
  k_scatter<N_OUT><<<blks((long long)E * (N_OUT / 4)), T, 0, stream>>>(AG, H, dis, srcv,
                                                                       dstv, E);

  // bias + log_softmax
  k_logsoftmax64<<<(n + 7) / 8, 256, 0, stream>>>(AG, b2, out, n);
}